// Resonator_13400297964157
// MI455X (gfx1250) — compile-verified
//
#include <hip/hip_runtime.h>
#include <math.h>

// ---------------------------------------------------------------------------
// Resonator for MI455X (gfx1250, wave32).
//
// Phases:
//  1) iir_partials : 1000 chunks x 480 samples, per-chunk IIR scan, zero carry
//  2) carry_scan   : single 1024-thread block, Hillis-Steele weighted prefix
//                    scan of chunk carries in LDS (weight r = (1-a)^480)
//  3) iir_apply    : re-scan each chunk with correct carry-in, write bandpass
//  4) taps_wmma    : 49-tap comb via V_WMMA_F32_16X16X4_F32, + tanh(gain*(..))
//
// Phase-4 notes:
//  * Powers feedback^tap computed by geometric recurrence (seed fb^(kh+1),
//    multiply by fb^4 per K-chunk) -- no transcendentals in the loop.
//  * Gathers are branch-free (clamped index + select) so EXEC stays all-1s
//    around every WMMA, per ISA 7.12 restrictions.
//  * Grid covers exactly N/16 tiles; no divergent early-out in the kernel.
//
// WMMA layout assumptions (compile-only environment, cannot validate):
//  A 16x4 f32 (documented, ISA 7.12.2): lane L holds row M=L%16;
//     v2f elem 0 -> VGPR0 -> K = (L<16 ? 0 : 2); elem 1 -> VGPR1 -> K+1.
//  B 4x16 f32 : assumed mirror split; B columns replicated (power per tap),
//     so only the A/B K-pairing matters; taken to match A's split.
//  D 16x16 f32 (documented): lane j<16 holds rows 0..7 col j in acc[0..7];
//     lane j>=16 holds rows 8..15 col j-16. Columns identical by construction,
//     so lane 0 emits out[base+0..7], lane 16 emits out[base+8..15].
// ---------------------------------------------------------------------------

typedef __attribute__((ext_vector_type(2))) float v2f;
typedef __attribute__((ext_vector_type(8))) float v8f;

#define NN      480000
#define CHUNK   480
#define NCHUNK  1000     // NN / CHUNK
#define SRF     48000.0f
#define MAXD    50

__device__ __forceinline__ float clampf(float x, float lo, float hi) {
    return fminf(fmaxf(x, lo), hi);
}

__device__ __forceinline__ void load_filter_coefs(const float* __restrict__ prm,
                                                  float& a_l, float& b_l,
                                                  float& a_h, float& b_h) {
    a_l = clampf(prm[2], 1e-7f, 0.99f);
    a_h = clampf(prm[3], 1e-7f, 0.99f);
    b_l = 1.0f - a_l;
    b_h = 1.0f - a_h;
}

// ---- Phase 1: per-chunk IIR partials (zero carry-in) ----------------------
__global__ void iir_partials(const float* __restrict__ buf,
                             const float* __restrict__ prm,
                             float* __restrict__ th,
                             float* __restrict__ tl) {
    int c = blockIdx.x * blockDim.x + threadIdx.x;
    if (c >= NCHUNK) return;
    float a_l, b_l, a_h, b_h;
    load_filter_coefs(prm, a_l, b_l, a_h, b_h);
    float yh = 0.0f, yl = 0.0f;
    const float* x = buf + c * CHUNK;
#pragma unroll 4
    for (int i = 0; i < CHUNK; ++i) {
        float v = x[i];
        yh = fmaf(b_h, yh, a_h * v);   // y = a*x + (1-a)*y
        yl = fmaf(b_l, yl, a_l * v);
    }
    th[c] = yh;
    tl[c] = yl;
}

// ---- Phase 2: weighted prefix scan of chunk carries (one block) -----------
// S[c] = r*S[c-1] + t[c]  ->  carry into chunk c is S[c-1].
__global__ void carry_scan(const float* __restrict__ prm,
                           const float* __restrict__ th,
                           const float* __restrict__ tl,
                           float* __restrict__ Sh,
                           float* __restrict__ Sl) {
    __shared__ float sh[1024];
    __shared__ float sl[1024];
    int t = threadIdx.x;
    float a_l, b_l, a_h, b_h;
    load_filter_coefs(prm, a_l, b_l, a_h, b_h);
    float rh = powf(b_h, (float)CHUNK);   // decay across one chunk
    float rl = powf(b_l, (float)CHUNK);

    sh[t] = (t < NCHUNK) ? th[t] : 0.0f;
    sl[t] = (t < NCHUNK) ? tl[t] : 0.0f;
    __syncthreads();

    for (int o = 1; o < 1024; o <<= 1) {
        float vh = (t >= o) ? sh[t - o] : 0.0f;
        float vl = (t >= o) ? sl[t - o] : 0.0f;
        __syncthreads();
        sh[t] = fmaf(rh, vh, sh[t]);
        sl[t] = fmaf(rl, vl, sl[t]);
        __syncthreads();
        rh *= rh;                          // r^o -> r^(2o)
        rl *= rl;
    }
    if (t < NCHUNK) { Sh[t] = sh[t]; Sl[t] = sl[t]; }
}

// ---- Phase 3: re-scan chunks with carry, write bandpass -------------------
__global__ void iir_apply(const float* __restrict__ buf,
                          const float* __restrict__ prm,
                          const float* __restrict__ Sh,
                          const float* __restrict__ Sl,
                          float* __restrict__ filt) {
    int c = blockIdx.x * blockDim.x + threadIdx.x;
    if (c >= NCHUNK) return;
    float a_l, b_l, a_h, b_h;
    load_filter_coefs(prm, a_l, b_l, a_h, b_h);
    float yh = (c > 0) ? Sh[c - 1] : 0.0f;
    float yl = (c > 0) ? Sl[c - 1] : 0.0f;
    const float* x = buf + c * CHUNK;
    float* f = filt + c * CHUNK;
#pragma unroll 4
    for (int i = 0; i < CHUNK; ++i) {
        float v = x[i];
        yh = fmaf(b_h, yh, a_h * v);
        yl = fmaf(b_l, yl, a_l * v);
        f[i] = yh - yl;                    // bandpass = lp(high) - lp(low)
    }
}

// ---- Phase 4: 49-tap comb via WMMA, then tanh -----------------------------
__global__ void __launch_bounds__(256)
taps_wmma(const float* __restrict__ in,
          const float* __restrict__ prm,
          const float* __restrict__ filt,
          float* __restrict__ out) {
    const int wave = blockIdx.x * (blockDim.x >> 5) + (threadIdx.x >> 5);
    const int base = wave * 16;            // grid covers exactly NN/16 tiles
    const int lane = threadIdx.x & 31;
    const int m    = lane & 15;            // A row (M) and D column index
    const int kh   = (lane >> 4) << 1;     // K half: lanes 0-15 -> {0,1}, 16-31 -> {2,3}

    // Device-side parameter derivation (matches reference clipping).
    float freq = clampf(prm[0], 20.0f, SRF * 0.5f);
    float fb   = clampf(prm[1], 0.0f, 0.99f);
    float gain = fmaxf(prm[4], 1e-7f);
    int DL = (int)floorf(SRF / freq);
    DL = min(max(DL, 1), NN / 2);
    int nd = min(NN / DL, MAXD);

    // Geometric power recurrence: this lane's first taps are kh+1 and kh+2;
    // taps advance by 4 per K-chunk -> multiply by fb^4.
    const float fb2 = fb * fb;
    const float fb4 = fb2 * fb2;
    float pw0 = (kh == 0) ? fb : fb * fb2;     // fb^(kh+1)
    float pw1 = pw0 * fb;                      // fb^(kh+2)

    v8f acc = {};
#pragma unroll
    for (int c = 0; c < 13; ++c) {         // taps 1..52 in chunks of 4 (50..52 masked)
        const int t0 = c * 4 + kh + 1;
        const int t1 = t0 + 1;
        // A: gathered filtered data, rows = 16 consecutive output samples.
        // Branch-free: clamp index, select zero afterwards (keeps EXEC all-1s).
        const int i0 = base + m - t0 * DL;
        const int i1 = base + m - t1 * DL;
        const float g0 = filt[max(i0, 0)];
        const float g1 = filt[max(i1, 0)];
        const float a0 = (i0 >= 0) ? g0 : 0.0f;
        const float a1 = (i1 >= 0) ? g1 : 0.0f;
        // B: feedback^tap replicated across N; inactive taps -> power 0.
        const float p0 = ((t0 < nd) && (t0 * DL < NN)) ? pw0 : 0.0f;
        const float p1 = ((t1 < nd) && (t1 * DL < NN)) ? pw1 : 0.0f;
        pw0 *= fb4;
        pw1 *= fb4;
        v2f A = {a0, a1};
        v2f B = {p0, p1};
        // (neg_a, A, neg_b, B, c_mod, C, reuse_a, reuse_b)
        acc = __builtin_amdgcn_wmma_f32_16x16x4_f32(false, A, false, B,
                                                    (short)0, acc, false, false);
    }

    // D columns identical: lane 0 carries rows 0..7, lane 16 carries rows 8..15.
    if ((lane & 15) == 0) {
        const int off = (lane >> 4) * 8;
#pragma unroll
        for (int r = 0; r < 8; ++r) {
            const int n = base + off + r;
            out[n] = tanhf(gain * (in[n] + acc[r]));
        }
    }
}

// ---------------------------------------------------------------------------
extern "C" void kernel_launch(void* const* d_in, const int* in_sizes, int n_in,
                              void* d_out, int out_size, void* d_ws, size_t ws_size,
                              hipStream_t stream) {
    (void)in_sizes; (void)n_in; (void)out_size; (void)ws_size;
    const float* input  = (const float*)d_in[0];   // [480000]
    const float* params = (const float*)d_in[1];   // [5]
    const float* dbuf   = (const float*)d_in[2];   // [480000]
    float* out = (float*)d_out;                    // [480000]

    float* ws   = (float*)d_ws;
    float* filt = ws;                 // NN floats
    float* th   = ws + NN;            // 1024
    float* tl   = th + 1024;          // 1024
    float* Sh   = tl + 1024;          // 1024
    float* Sl   = Sh + 1024;          // 1024

    // Phase 1: per-chunk partial IIR scans (1000 chunks).
    iir_partials<<<(NCHUNK + 255) / 256, 256, 0, stream>>>(dbuf, params, th, tl);
    // Phase 2: cross-chunk carry scan (single block, 10 LDS steps).
    carry_scan<<<1, 1024, 0, stream>>>(params, th, tl, Sh, Sl);
    // Phase 3: apply carries, write bandpass-filtered signal.
    iir_apply<<<(NCHUNK + 255) / 256, 256, 0, stream>>>(dbuf, params, Sh, Sl, filt);
    // Phase 4: WMMA tap accumulation + tanh. 30000 tiles of 16 samples,
    // 8 waves (256 threads) per block -> 3750 blocks (exact cover).
    const int tiles = NN / 16;
    taps_wmma<<<tiles / 8, 256, 0, stream>>>(input, params, filt, out);
}